// TimeReasoning_81037442941612
// MI455X (gfx1250) — compile-verified
//
#include <hip/hip_runtime.h>
#include <hip/hip_bf16.h>
#include <math.h>

// Problem constants (from reference)
#define Bc   32
#define Tc   16
#define Sc   128
#define Kc   64
#define Hc   512
#define MIDc 512
#define Ec   64
#define Vc   2048
#define NST  15   // T-1 scan steps

typedef float v2f __attribute__((ext_vector_type(2)));
typedef float v8f __attribute__((ext_vector_type(8)));

// D = A(16x4 f32) x B(4x16 f32) + C(16x16 f32), wave32 WMMA
__device__ __forceinline__ v8f wmma4(v2f a, v2f b, v8f c) {
  return __builtin_amdgcn_wmma_f32_16x16x4_f32(false, a, false, b, (short)0, c, false, false);
}

// ---------------------------------------------------------------------------
// Kernel 1: post = sum_k logits (with -inf -> 0), act = softmax(post),
//           z = softmax(post + g1), lp1 = sum z * log_softmax(post)
// grid: 15*32 blocks (t*32+b), 128 threads (s)
// ---------------------------------------------------------------------------
__global__ void k_post(const float* __restrict__ logits, const float* __restrict__ g1,
                       float* __restrict__ post, float* __restrict__ act,
                       float* __restrict__ z, float* __restrict__ lp1) {
  int blk = blockIdx.x;            // t*32 + b
  int t = blk >> 5, b = blk & 31;
  int s = threadIdx.x;
  __shared__ float red[Sc];
  const float* lp = logits + ((size_t)((b * Tc + t) * Sc + s)) * Kc;
  float p = 0.f;
  for (int k = 0; k < Kc; ++k) {
    float v = lp[k];
    p += (v == -INFINITY) ? 0.f : v;
  }
  post[blk * Sc + s] = p;
  // softmax over s (act) + logsumexp
  red[s] = p; __syncthreads();
  for (int off = 64; off > 0; off >>= 1) { if (s < off) red[s] = fmaxf(red[s], red[s + off]); __syncthreads(); }
  float m = red[0]; __syncthreads();
  float e = __expf(p - m);
  red[s] = e; __syncthreads();
  for (int off = 64; off > 0; off >>= 1) { if (s < off) red[s] += red[s + off]; __syncthreads(); }
  float sum = red[0]; __syncthreads();
  float lse = m + __logf(sum);
  act[blk * Sc + s] = e / sum;
  // z = softmax(post + g1)
  float u = p + g1[blk * Sc + s];
  red[s] = u; __syncthreads();
  for (int off = 64; off > 0; off >>= 1) { if (s < off) red[s] = fmaxf(red[s], red[s + off]); __syncthreads(); }
  float mz = red[0]; __syncthreads();
  float ez = __expf(u - mz);
  red[s] = ez; __syncthreads();
  for (int off = 64; off > 0; off >>= 1) { if (s < off) red[s] += red[s + off]; __syncthreads(); }
  float sz = red[0]; __syncthreads();
  float zv = ez / sz;
  z[blk * Sc + s] = zv;
  // lp1 = sum_s z * (post - lse)
  red[s] = zv * (p - lse); __syncthreads();
  for (int off = 64; off > 0; off >>= 1) { if (s < off) red[s] += red[s + off]; __syncthreads(); }
  if (s == 0) lp1[blk] = red[0];
}

// ---------------------------------------------------------------------------
// Kernel 2: qw[t,b,e] = relu(q_emb[t,b,:] @ Wq + bq)[e] * Wout[e]; t==15 -> 0
// grid: 16*32 blocks, 64 threads (e)
// ---------------------------------------------------------------------------
__global__ void k_qe(const float* __restrict__ q_emb, const float* __restrict__ Wq,
                     const float* __restrict__ bq, const float* __restrict__ Wout,
                     float* __restrict__ qw) {
  int blk = blockIdx.x;            // t*32 + b
  int t = blk >> 5, b = blk & 31;
  int e = threadIdx.x;
  if (t == NST) { qw[blk * Ec + e] = 0.f; return; }  // zero pad column (whole block)
  __shared__ float qs[MIDc];
  for (int m = e; m < MIDc; m += 64) qs[m] = q_emb[(t * Bc + b) * MIDc + m];
  __syncthreads();
  float acc = bq[e];
  for (int m = 0; m < MIDc; ++m) acc += qs[m] * Wq[m * Ec + e];
  acc = fmaxf(acc, 0.f);
  qw[blk * Ec + e] = acc * Wout[e];
}

// ---------------------------------------------------------------------------
// Kernel 3: h_act[t,b,h] = sum_s object[b,s,h] * act[t,b,s]
// grid: 15*32 blocks, 256 threads
// ---------------------------------------------------------------------------
__global__ void k_hact(const float* __restrict__ obj, const float* __restrict__ act,
                       float* __restrict__ hact) {
  int blk = blockIdx.x;            // t*32 + b
  int b = blk & 31;
  int tid = threadIdx.x;
  __shared__ float as[Sc];
  if (tid < Sc) as[tid] = act[blk * Sc + tid];
  __syncthreads();
  for (int h = tid; h < Hc; h += 256) {
    float acc = 0.f;
    for (int s = 0; s < Sc; ++s) acc += obj[(size_t)(b * Sc + s) * Hc + h] * as[s];
    hact[blk * Hc + h] = acc;
  }
}

// ---------------------------------------------------------------------------
// Kernel 4 (WMMA): per (b, j-tile of 16):
//   rel[t,i,jl] = relu(edge[b,i,j0+jl,:] . qw[t,b,:] + bout)   via f32 WMMA
//   softmax over i (with r_mask) per (jl,t) column, then partial
//   new_z part[b,jt,t,i] = sum_jl rel_m[i,jl,t] * post[t,b,j0+jl]
// grid: 32*8 blocks, 256 threads (8 waves)
// Double-buffered A fragments: 16-load clause in flight during 16-WMMA chain.
// ---------------------------------------------------------------------------
__global__ void k_rel(const float* __restrict__ edge, const float* __restrict__ r_mask,
                      const float* __restrict__ qw, const float* __restrict__ post,
                      const float* __restrict__ boutp, float* __restrict__ nzpart) {
  __shared__ float relS[Sc * 256];    // [i][jl*16 + t]  (128 KB)
  __shared__ float maskS[Sc * 16];    // [i][jl]
  __shared__ float postS[NST][16];    // [t][jl]
  int blk = blockIdx.x;               // b*8 + jt
  int b = blk >> 3, jt = blk & 7;
  int j0 = jt * 16;
  int tid = threadIdx.x;
  int wave = tid >> 5, lane = tid & 31;
  int half = lane >> 4, n = lane & 15;

  for (int idx = tid; idx < Sc * 16; idx += 256) {
    int i = idx >> 4, jl = idx & 15;
    maskS[idx] = r_mask[(b * Sc + i) * Sc + (j0 + jl)];
  }
  for (int idx = tid; idx < NST * 16; idx += 256) {
    int t = idx >> 4, jl = idx & 15;
    postS[t][jl] = post[(t * Bc + b) * Sc + (j0 + jl)];
  }
  float bout = boutp[0];

  // Preload B fragments: B[k=e][N=t] = qw[t,b,e]; reused for all 128 i's.
  v2f bfrag[16];
#pragma unroll
  for (int ks = 0; ks < 16; ++ks) {
    int e0 = 4 * ks + 2 * half;
    bfrag[ks] = *reinterpret_cast<const v2f*>(qw + (n * Bc + b) * Ec + e0);
  }

  // A-row fragment loader (per-lane: row j0+n, K-pairs at stride 4 floats)
  const float* edgeLane = edge + ((size_t)b * Sc * Sc + (j0 + n)) * Ec + 2 * half;
  auto aload = [&](v2f* dst, int i) {
#pragma unroll
    for (int ks = 0; ks < 16; ++ks)
      dst[ks] = *reinterpret_cast<const v2f*>(edgeLane + (size_t)i * Sc * Ec + 4 * ks);
    if (i + 2 < Sc)
      __builtin_prefetch(edgeLane + (size_t)(i + 2) * Sc * Ec, 0, 0);
  };
  auto compute = [&](const v2f* a, int i) {
    v8f acc;
#pragma unroll
    for (int r = 0; r < 8; ++r) acc[r] = bout;
#pragma unroll
    for (int ks = 0; ks < 16; ++ks) acc = wmma4(a[ks], bfrag[ks], acc);
#pragma unroll
    for (int r = 0; r < 8; ++r) {
      int jl = r + 8 * half;                       // D: M = r + 8*half, N = n
      relS[i * 256 + jl * 16 + n] = fmaxf(acc[r], 0.f);
    }
  };

  // Phase 1: software-pipelined WMMA over the wave's 16 rows (i)
  const int iBase = wave * 16;
  v2f aA[16], aB[16];
  aload(aA, iBase);
#pragma unroll
  for (int ii = 0; ii < 16; ii += 2) {
    aload(aB, iBase + ii + 1);                     // in flight during compute
    compute(aA, iBase + ii);
    if (ii + 2 < 16) aload(aA, iBase + ii + 2);
    compute(aB, iBase + ii + 1);
  }
  __syncthreads();

  // Phase 2: softmax over i for each column c = jl*16 + t (in-place normalize)
  {
    int c = tid, jl = c >> 4;
    float m = -3.4e38f;
    for (int i = 0; i < Sc; ++i) {
      float v = (maskS[i * 16 + jl] == 0.f) ? -1e12f : relS[i * 256 + c];
      m = fmaxf(m, v);
    }
    float s = 0.f;
    for (int i = 0; i < Sc; ++i) {
      float v = (maskS[i * 16 + jl] == 0.f) ? -1e12f : relS[i * 256 + c];
      s += __expf(v - m);
    }
    float inv = 1.f / s;
    for (int i = 0; i < Sc; ++i) {
      float v = (maskS[i * 16 + jl] == 0.f) ? -1e12f : relS[i * 256 + c];
      relS[i * 256 + c] = __expf(v - m) * inv;
    }
  }
  __syncthreads();

  // Phase 3: partial new_z over this j-tile
  for (int idx = tid; idx < 16 * Sc; idx += 256) {
    int t = idx & 15;
    if (t == NST) continue;
    int i = idx >> 4;
    float acc = 0.f;
#pragma unroll
    for (int jl = 0; jl < 16; ++jl)
      acc += relS[i * 256 + jl * 16 + t] * postS[t][jl];
    nzpart[((b * 8 + jt) * 16 + t) * Sc + i] = acc;
  }
}

// ---------------------------------------------------------------------------
// Kernel 5 (WMMA): recons = h_act(480x512) @ Wd(512x2048) + bd, fused BCE vs
// e_label, row-partial mean -> rpart[row][nblk]
// grid: 30*8 blocks (Mtile*8+nblk), 256 threads (8 waves x 2 N-tiles)
// K processed in chunks of 8: one load clause (8xb64 + 16 scalar), 8 WMMAs.
// ---------------------------------------------------------------------------
__global__ void k_recon(const float* __restrict__ hact, const float* __restrict__ Wd,
                        const float* __restrict__ bd, const float* __restrict__ elab,
                        float* __restrict__ rpart) {
  int blk = blockIdx.x;
  int Mtile = blk >> 3, nblk = blk & 7;
  int row0 = Mtile * 16, col0 = nblk * 256;
  int tid = threadIdx.x, wave = tid >> 5, lane = tid & 31;
  int half = lane >> 4, n = lane & 15;
  __shared__ float wsumS[8][16];
  float rAcc[8];
#pragma unroll
  for (int r = 0; r < 8; ++r) rAcc[r] = 0.f;
  int m = row0 + n;                                // A row (M = lane%16)
  const float* hrow = hact + m * Hc + 2 * half;
  for (int tt = 0; tt < 2; ++tt) {
    int colT = col0 + (wave * 2 + tt) * 16;
    int v = colT + n;                              // B col (N = lane%16)
    float bdv = bd[v];
    v8f acc;
#pragma unroll
    for (int r = 0; r < 8; ++r) acc[r] = bdv;
    for (int kc = 0; kc < 128; kc += 8) {
      v2f af[8], bf[8];
#pragma unroll
      for (int q = 0; q < 8; ++q) {
        int e0 = 4 * (kc + q) + 2 * half;
        af[q] = *reinterpret_cast<const v2f*>(hrow + 4 * (kc + q));
        bf[q][0] = Wd[e0 * Vc + v];
        bf[q][1] = Wd[(e0 + 1) * Vc + v];
      }
#pragma unroll
      for (int q = 0; q < 8; ++q) acc = wmma4(af[q], bf[q], acc);
    }
#pragma unroll
    for (int r = 0; r < 8; ++r) {
      int row = row0 + r + 8 * half;               // D: M = r + 8*half
      int t = row >> 5, bb = row & 31;
      float x = acc[r];
      float y = elab[(size_t)(bb * Tc + t) * Vc + v];
      rAcc[r] += fmaxf(x, 0.f) - x * y + log1pf(__expf(-fabsf(x)));
    }
  }
  // reduce over the 16 lanes of each half (fixed-order butterfly: deterministic)
#pragma unroll
  for (int r = 0; r < 8; ++r) {
    float vs = rAcc[r];
    for (int off = 1; off < 16; off <<= 1) vs += __shfl_xor(vs, off, 32);
    rAcc[r] = vs;
  }
  if (n == 0) {
#pragma unroll
    for (int r = 0; r < 8; ++r) wsumS[wave][r + 8 * half] = rAcc[r];
  }
  __syncthreads();
  if (tid < 16) {
    float s = 0.f;
#pragma unroll
    for (int w = 0; w < 8; ++w) s += wsumS[w][tid];
    rpart[(row0 + tid) * 8 + nblk] = s;
  }
}

// ---------------------------------------------------------------------------
// Kernel 6: new_z[t-1,b,i] = sum_jt part; lp2[t,b] = sum_i z[t]*log_softmax(new_z)
// grid: 14*32 blocks ((t-1)*32+b, t=1..14), 128 threads (i)
// ---------------------------------------------------------------------------
__global__ void k_nz(const float* __restrict__ nzpart, const float* __restrict__ z,
                     float* __restrict__ lp2) {
  int blk = blockIdx.x;
  int tm1 = blk >> 5, b = blk & 31;
  int t = tm1 + 1;
  int i = threadIdx.x;
  __shared__ float red[Sc];
  float nz = 0.f;
#pragma unroll
  for (int jt = 0; jt < 8; ++jt) nz += nzpart[((b * 8 + jt) * 16 + tm1) * Sc + i];
  red[i] = nz; __syncthreads();
  for (int off = 64; off > 0; off >>= 1) { if (i < off) red[i] = fmaxf(red[i], red[i + off]); __syncthreads(); }
  float m = red[0]; __syncthreads();
  red[i] = __expf(nz - m); __syncthreads();
  for (int off = 64; off > 0; off >>= 1) { if (i < off) red[i] += red[i + off]; __syncthreads(); }
  float lse = m + __logf(red[0]); __syncthreads();
  float zi = z[(t * Bc + b) * Sc + i];
  red[i] = zi * (nz - lse); __syncthreads();
  for (int off = 64; off > 0; off >>= 1) { if (i < off) red[i] += red[i + off]; __syncthreads(); }
  if (i == 0) lp2[t * Bc + b] = red[0];
}

// ---------------------------------------------------------------------------
// Kernel 7: final scalar = mean_b sum_t (recon + (t>0)*(lp1-lp2)) * e_mask
// 1 block, 512 threads
// ---------------------------------------------------------------------------
__global__ void k_fin(const float* __restrict__ rpart, const float* __restrict__ lp1,
                      const float* __restrict__ lp2, const float* __restrict__ emask,
                      float* __restrict__ out) {
  __shared__ float red[512];
  int tid = threadIdx.x;
  float val = 0.f;
  if (tid < NST * Bc) {
    int row = tid;                   // t*32 + b
    int t = row >> 5, b = row & 31;
    float rs = 0.f;
#pragma unroll
    for (int nb = 0; nb < 8; ++nb) rs += rpart[row * 8 + nb];
    float l = rs * (1.f / (float)Vc);
    if (t > 0) l += lp1[row] - lp2[row];
    val = l * emask[b * Tc + t];
  }
  red[tid] = val; __syncthreads();
  for (int off = 256; off > 0; off >>= 1) { if (tid < off) red[tid] += red[tid + off]; __syncthreads(); }
  if (tid == 0) out[0] = red[0] / (float)Bc;
}

// ---------------------------------------------------------------------------
extern "C" void kernel_launch(void* const* d_in, const int* in_sizes, int n_in,
                              void* d_out, int out_size, void* d_ws, size_t ws_size,
                              hipStream_t stream) {
  const float* q_emb  = (const float*)d_in[0];
  const float* obj    = (const float*)d_in[1];
  const float* elab   = (const float*)d_in[2];
  // d_in[3]=e_s, d_in[4]=e_e (unused by reference)
  const float* emask  = (const float*)d_in[5];
  const float* edge   = (const float*)d_in[6];
  const float* rmask  = (const float*)d_in[7];
  const float* logits = (const float*)d_in[8];
  const float* Wq     = (const float*)d_in[9];
  const float* bq     = (const float*)d_in[10];
  const float* Wout   = (const float*)d_in[11];
  const float* bout   = (const float*)d_in[12];
  const float* Wd     = (const float*)d_in[13];
  const float* bd     = (const float*)d_in[14];
  const float* g1     = (const float*)d_in[15];
  // d_in[16]=g2 (unused: l_back is dead code in the reference)

  // Workspace layout (floats), total ~3.8 MB
  float* w     = (float*)d_ws;
  float* post  = w;                       // 15*32*128
  float* act   = post + NST * Bc * Sc;    // 15*32*128
  float* z     = act  + NST * Bc * Sc;    // 15*32*128
  float* qw    = z    + NST * Bc * Sc;    // 16*32*64 (t=15 zero pad)
  float* hact  = qw   + Tc * Bc * Ec;     // 480*512
  float* lp1   = hact + NST * Bc * Hc;    // 480
  float* lp2   = lp1  + NST * Bc;         // 480
  float* rpart = lp2  + NST * Bc;         // 480*8
  float* nzp   = rpart + NST * Bc * 8;    // 32*8*16*128

  k_post <<<dim3(NST * Bc), dim3(Sc),  0, stream>>>(logits, g1, post, act, z, lp1);
  k_qe   <<<dim3(Tc * Bc),  dim3(Ec),  0, stream>>>(q_emb, Wq, bq, Wout, qw);
  k_hact <<<dim3(NST * Bc), dim3(256), 0, stream>>>(obj, act, hact);
  k_rel  <<<dim3(Bc * 8),   dim3(256), 0, stream>>>(edge, rmask, qw, post, bout, nzp);
  k_recon<<<dim3(30 * 8),   dim3(256), 0, stream>>>(hact, Wd, bd, elab, rpart);
  k_nz   <<<dim3(14 * Bc),  dim3(Sc),  0, stream>>>(nzp, z, lp2);
  k_fin  <<<dim3(1),        dim3(512), 0, stream>>>(rpart, lp1, lp2, emask, (float*)d_out);
}